// HashGrid_5763846111699
// MI455X (gfx1250) — compile-verified
//
#include <hip/hip_runtime.h>
#include <stdint.h>

// ---------------------------------------------------------------------------
// Instant-NGP hash-grid encoding for MI455X (gfx1250, wave32).
//
// Roofline: 256 MB output stream (NT stores), 64 MB of hash tables resident
// in the 192 MB L2 (gathers are the bottleneck -> 8-wide MLP per level,
// 16 levels fully unrolled). Input x tile staged block-wide into LDS with the
// Tensor Data Mover (tensor_load_to_lds + s_wait_tensorcnt), which is the
// CDNA5-specific data-movement path for this op (WMMA does not apply: no
// shared matrix operand exists in a per-point gather/interpolation).
// ---------------------------------------------------------------------------

typedef unsigned int u32x4 __attribute__((ext_vector_type(4)));
typedef int          i32x8 __attribute__((ext_vector_type(8)));
typedef int          i32x4 __attribute__((ext_vector_type(4)));
typedef float        f32x4 __attribute__((ext_vector_type(4)));
typedef float        f32x2 __attribute__((ext_vector_type(2)));

#define HG_L      16
#define HG_LOGT   19
#define HG_T      (1u << HG_LOGT)
#define HG_MASK   (HG_T - 1u)
#define HG_BLOCK  256
#define HG_F      2

__global__ __launch_bounds__(HG_BLOCK)
void hashgrid_encode_kernel(const float* __restrict__ xin,
                            const float* __restrict__ table,
                            float* __restrict__ out,
                            int npts) {
  __shared__ float sx[HG_BLOCK * 3];

  const int tid = threadIdx.x;
  const int p   = blockIdx.x * HG_BLOCK + tid;

#if __has_builtin(__builtin_amdgcn_tensor_load_to_lds) && __has_builtin(__builtin_amdgcn_s_wait_tensorcnt)
  // Wave 0 issues one TDM descriptor that DMAs this block's x-tile
  // (256 points * 3 floats = 768 dwords) from global into LDS.
  if (tid < 32) {
    const uint64_t base_elem = (uint64_t)blockIdx.x * (HG_BLOCK * 3u);
    const uint64_t ga = (uint64_t)(uintptr_t)xin + base_elem * 4ull;
    const uint32_t total_dw = (uint32_t)npts * 3u;
    const uint32_t remain   = total_dw - (uint32_t)base_elem;
    const uint32_t dim0     = remain < 768u ? remain : 768u;  // OOB reads -> 0

    // D# group 0: count=1 | lds_addr | global_addr[56:0] | type=2
    u32x4 g0;
    g0[0] = 1u;                                         // count=1, user mode
    g0[1] = 0u;                                         // lds_addr = 0 (sx[])
    g0[2] = (uint32_t)(ga & 0xFFFFFFFFull);             // global_addr[31:0]
    g0[3] = (uint32_t)((ga >> 32) & 0x1FFFFFFull)       // global_addr[56:32]
          | 0x80000000u;                                // type = 2 ("image")

    // D# group 1: data_size=4B, 1-row 2D tile of 768 dwords.
    i32x8 g1;
    g1[0] = (int)(2u << 16);                            // data_size code 2 = 4B
    g1[1] = (int)((dim0 & 0xFFFFu) << 16);              // tensor_dim0[15:0]
    g1[2] = (int)(((dim0 >> 16) & 0xFFFFu)              // tensor_dim0[31:16]
          | (1u << 16));                                // tensor_dim1 = 1
    g1[3] = (int)(768u << 16);                          // tile_dim0 = 768
    g1[4] = 1;                                          // tile_dim1=1, tile_dim2 unused
    g1[5] = 768;                                        // tensor_dim0_stride[31:0]
    g1[6] = 0;
    g1[7] = 0;
    i32x4 gz4 = {0, 0, 0, 0};                           // groups 2/3 unused (<=2D)
    i32x8 gz8 = {0, 0, 0, 0, 0, 0, 0, 0};

    __builtin_amdgcn_tensor_load_to_lds(g0, g1, gz4, gz4, gz8, 0);
    __builtin_amdgcn_s_wait_tensorcnt(0);               // wave0: DMA complete
  }
  __syncthreads();                                      // publish LDS to all waves
  // stride-3 dword addressing: 3 coprime 64 banks -> conflict-free ds_load_b32
  const float px = sx[tid * 3 + 0];
  const float py = sx[tid * 3 + 1];
  const float pz = sx[tid * 3 + 2];
#else
  const int pr = p < npts ? p : 0;
  const float px = xin[pr * 3 + 0];
  const float py = xin[pr * 3 + 1];
  const float pz = xin[pr * 3 + 2];
#endif

  // Per-level resolutions: int(16 * 2^(l/3)), l = 0..15.
  constexpr float RESF[HG_L] = {16.f, 20.f, 25.f, 32.f, 40.f, 50.f, 64.f, 80.f,
                                101.f, 128.f, 161.f, 203.f, 256.f, 322.f, 406.f, 512.f};

  float acc[2 * HG_L];

#pragma unroll
  for (int l = 0; l < HG_L; ++l) {
    const float r   = RESF[l];
    const float sxv = px * r, syv = py * r, szv = pz * r;
    const float fx = floorf(sxv), fy = floorf(syv), fz = floorf(szv);
    const float wx1 = sxv - fx, wy1 = syv - fy, wz1 = szv - fz;
    const float wx0 = 1.f - wx1, wy0 = 1.f - wy1, wz0 = 1.f - wz1;

    const uint32_t cx = (uint32_t)(int)fx;
    const uint32_t cy = (uint32_t)(int)fy;
    const uint32_t cz = (uint32_t)(int)fz;

    // spatial hash h = x*1 ^ y*2654435761 ^ z*805459861  (uint32 wraparound)
    const uint32_t hx0 = cx,                     hx1 = cx + 1u;
    const uint32_t hy0 = cy * 2654435761u,       hy1 = (cy + 1u) * 2654435761u;
    const uint32_t hz0 = cz * 805459861u,        hz1 = (cz + 1u) * 805459861u;

    const uint32_t i0 = (hx0 ^ hy0 ^ hz0) & HG_MASK;   // corner (0,0,0)
    const uint32_t i1 = (hx0 ^ hy0 ^ hz1) & HG_MASK;   // (0,0,1)
    const uint32_t i2 = (hx0 ^ hy1 ^ hz0) & HG_MASK;   // (0,1,0)
    const uint32_t i3 = (hx0 ^ hy1 ^ hz1) & HG_MASK;   // (0,1,1)
    const uint32_t i4 = (hx1 ^ hy0 ^ hz0) & HG_MASK;   // (1,0,0)
    const uint32_t i5 = (hx1 ^ hy0 ^ hz1) & HG_MASK;   // (1,0,1)
    const uint32_t i6 = (hx1 ^ hy1 ^ hz0) & HG_MASK;   // (1,1,0)
    const uint32_t i7 = (hx1 ^ hy1 ^ hz1) & HG_MASK;   // (1,1,1)

    // 8 independent b64 gathers (L2-resident tables) -> high MLP.
    const f32x2* __restrict__ tbl = (const f32x2*)table + ((size_t)l << HG_LOGT);
    const f32x2 f0 = tbl[i0];
    const f32x2 f1 = tbl[i1];
    const f32x2 f2 = tbl[i2];
    const f32x2 f3 = tbl[i3];
    const f32x2 f4 = tbl[i4];
    const f32x2 f5 = tbl[i5];
    const f32x2 f6 = tbl[i6];
    const f32x2 f7 = tbl[i7];

    // trilinear weights, same ((wx*wy)*wz) association as the reference
    const float wxy00 = wx0 * wy0, wxy01 = wx0 * wy1;
    const float wxy10 = wx1 * wy0, wxy11 = wx1 * wy1;

    float w, a0, a1;
    w = wxy00 * wz0; a0 = w * f0.x;             a1 = w * f0.y;
    w = wxy00 * wz1; a0 = fmaf(w, f1.x, a0);    a1 = fmaf(w, f1.y, a1);
    w = wxy01 * wz0; a0 = fmaf(w, f2.x, a0);    a1 = fmaf(w, f2.y, a1);
    w = wxy01 * wz1; a0 = fmaf(w, f3.x, a0);    a1 = fmaf(w, f3.y, a1);
    w = wxy10 * wz0; a0 = fmaf(w, f4.x, a0);    a1 = fmaf(w, f4.y, a1);
    w = wxy10 * wz1; a0 = fmaf(w, f5.x, a0);    a1 = fmaf(w, f5.y, a1);
    w = wxy11 * wz0; a0 = fmaf(w, f6.x, a0);    a1 = fmaf(w, f6.y, a1);
    w = wxy11 * wz1; a0 = fmaf(w, f7.x, a0);    a1 = fmaf(w, f7.y, a1);

    acc[2 * l + 0] = a0;
    acc[2 * l + 1] = a1;
  }

  if (p < npts) {
    // Each lane owns its full 128B output row: 8 consecutive NT b128 stores
    // -> GL2 write-combines whole lines; NT keeps the 64MB of hash tables
    // resident in L2 instead of being evicted by the 256MB output stream.
    f32x4* __restrict__ o = (f32x4*)(out + (size_t)p * (2 * HG_L));
#pragma unroll
    for (int j = 0; j < 8; ++j) {
      f32x4 v;
      v.x = acc[4 * j + 0];
      v.y = acc[4 * j + 1];
      v.z = acc[4 * j + 2];
      v.w = acc[4 * j + 3];
      __builtin_nontemporal_store(v, o + j);
    }
  }
}

extern "C" void kernel_launch(void* const* d_in, const int* in_sizes, int n_in,
                              void* d_out, int out_size, void* d_ws, size_t ws_size,
                              hipStream_t stream) {
  const float* x       = (const float*)d_in[0];   // [P,3] f32
  const float* hashmap = (const float*)d_in[1];   // [L,T,F] f32
  float*       out     = (float*)d_out;           // [P, L*F] f32

  const int npts   = in_sizes[0] / 3;
  const int blocks = (npts + HG_BLOCK - 1) / HG_BLOCK;

  hipLaunchKernelGGL(hashgrid_encode_kernel, dim3(blocks), dim3(HG_BLOCK), 0, stream,
                     x, hashmap, out, npts);
}